// AdaptiveSample_58926951301812
// MI455X (gfx1250) — compile-verified
//
#include <hip/hip_runtime.h>
#include <hip/hip_bf16.h>
#include <stdint.h>

#define K_SIZE     5
#define SAMPLE_NUM 15
#define DEPTH_MAXF 192.0f
#define BB 2
#define CC 32
#define HH 240
#define WW 320
#define TH 8
#define TW 32
#define RROWS (TH + 4)            // 12 halo rows
#define RCOLS (TW + 4)            // 36 halo cols
#define NPIX  (RROWS * RCOLS)     // 432
#define FSTRIDE 36                // floats per pixel channel-block (144B: 16B-aligned, bank-conflict-free)
#define FEAT_FLOATS (NPIX * FSTRIDE)  // 15552
#define NORM_FLOATS (NPIX * 4)        // 1728
#define VAL_FLOATS  (NPIX)            // 432
#define SMEM_BYTES  ((FEAT_FLOATS + NORM_FLOATS + VAL_FLOATS) * 4 + 48 * 4)

typedef float vfloat4 __attribute__((ext_vector_type(4)));   // native vector: OK for nontemporal builtins

__device__ __forceinline__ int iclamp(int v, int lo, int hi) {
    return v < lo ? lo : (v > hi ? hi : v);
}

__device__ __forceinline__ void async_load_f32_to_lds(unsigned lds_byte_off, const float* gptr) {
    // CDNA5 async global->LDS DMA (per-lane LDS scatter address), tracked by ASYNCcnt.
    // Ordering vs. LDS readers is established by s_wait_asynccnt + s_barrier below,
    // so no memory clobber here (lets the scheduler interleave issue).
    asm volatile("global_load_async_to_lds_b32 %0, %1, off"
                 :: "v"(lds_byte_off), "v"((unsigned long long)(uintptr_t)gptr));
}

__global__ __launch_bounds__(256) void
AdaptiveSample_main(const float* __restrict__ depth,
                    const float* __restrict__ snorm,
                    const float* __restrict__ feat,
                    const float* __restrict__ guide,
                    const int*   __restrict__ sidx,
                    float* __restrict__ out) {
    extern __shared__ float smem[];
    float* sF = smem;                        // [NPIX][FSTRIDE] pixel-major, channels contiguous
    float* sN = smem + FEAT_FLOATS;          // [NPIX][4] normals (xyz + pad)
    float* sV = sN + NORM_FLOATS;            // [NPIX] validity
    int*   sP  = (int*)(sV + VAL_FLOATS);    // [16] raw sample positions
    int*   sDy = sP + 16;                    // [16]
    int*   sDx = sDy + 16;                   // [16]

    const int tid  = threadIdx.x;
    const int lane = tid & 31;
    const int wv   = tid >> 5;               // 8 waves
    const int b  = blockIdx.z;
    const int h0 = blockIdx.y * TH;
    const int w0 = blockIdx.x * TW;

    if (tid < SAMPLE_NUM) {
        int p = sidx[tid];
        sP[tid]  = p;
        sDy[tid] = p / K_SIZE - 2;
        sDx[tid] = p % K_SIZE - 2;
    }

    const float* featB = feat  + (size_t)b * CC * HH * WW;
    const float* snB   = snorm + (size_t)b * 3  * HH * WW;
    const float* depB  = depth + (size_t)b * HH * WW;

    // Lane-invariant column geometry (hoisted out of all staging loops).
    const int  gx1   = w0 - 2 + lane;               // x = lane
    const int  gx2   = gx1 + 32;                    // x = lane + 32 (lanes 0..3 only)
    const bool col1  = (unsigned)gx1 < (unsigned)WW;
    const bool col2  = (unsigned)gx2 < (unsigned)WW;
    const int  gx1c  = iclamp(gx1, 0, WW - 1);
    const int  gx2c  = iclamp(gx2, 0, WW - 1);

    // ---- Stage features (transposed to pixel-major) via async DMA; OOB -> exact zero ----
    // One (c,y) row of 36 elements per wave-iteration: 384 rows / 8 waves = 48 iters.
    for (int r = wv; r < CC * RROWS; r += 8) {
        const int c = r / RROWS;
        const int y = r - c * RROWS;
        const int gy = h0 - 2 + y;
        const bool rowIn = (unsigned)gy < (unsigned)HH;
        const float* gbase = featB + (size_t)c * (HH * WW) + (ptrdiff_t)gy * WW;
        const int idx1 = (y * RCOLS + lane) * FSTRIDE + c;          // x = lane
        if (rowIn && col1) async_load_f32_to_lds((unsigned)(uintptr_t)&sF[idx1], gbase + gx1);
        else               sF[idx1] = 0.0f;
        if (lane < 4) {
            const int idx2 = idx1 + 32 * FSTRIDE;                   // x = lane + 32
            if (rowIn && col2) async_load_f32_to_lds((unsigned)(uintptr_t)&sF[idx2], gbase + gx2);
            else               sF[idx2] = 0.0f;
        }
    }

    // ---- Stage normals (clamped coords: validity=0 already kills OOB samples) ----
    for (int r = wv; r < 3 * RROWS; r += 8) {                       // 36 rows
        const int c = r / RROWS;
        const int y = r - c * RROWS;
        const int gyc = iclamp(h0 - 2 + y, 0, HH - 1);
        const float* gbase = snB + (size_t)c * (HH * WW) + (size_t)gyc * WW;
        const int idx1 = (y * RCOLS + lane) * 4 + c;
        async_load_f32_to_lds((unsigned)(uintptr_t)&sN[idx1], gbase + gx1c);
        if (lane < 4)
            async_load_f32_to_lds((unsigned)(uintptr_t)&sN[idx1 + 32 * 4], gbase + gx2c);
    }

    // ---- Validity mask from depth (computed value, so regular loads + ds stores) ----
    for (int y = wv; y < RROWS; y += 8) {
        const int gy = h0 - 2 + y;
        const bool rowIn = (unsigned)gy < (unsigned)HH;
        const float* dbase = depB + (ptrdiff_t)gy * WW;
        float v1 = 0.0f;
        if (rowIn && col1) {
            float d = dbase[gx1];
            v1 = (d > 0.0f && d < DEPTH_MAXF) ? 1.0f : 0.0f;
        }
        sV[y * RCOLS + lane] = v1;
        if (lane < 4) {
            float v2 = 0.0f;
            if (rowIn && col2) {
                float d = dbase[gx2];
                v2 = (d > 0.0f && d < DEPTH_MAXF) ? 1.0f : 0.0f;
            }
            sV[y * RCOLS + lane + 32] = v2;
        }
    }

    asm volatile("s_wait_asynccnt 0" ::: "memory");   // drain this wave's async DMA
    __syncthreads();

    // ---- Per-pixel compute ----
    const int tx = lane;
    const int ty = wv;
    const int h = h0 + ty;
    const int w = w0 + tx;
    const int lr = ty + 2;
    const int lc = tx + 2;
    const size_t pix = (size_t)h * WW + w;

    // center normal: reference uses a raw reshape [B,3,H,W]->[B,H,W,3] (NOT a transpose)
    const float* cn = snorm + (size_t)b * 3 * HH * WW + pix * 3;
    const float cnx = cn[0], cny = cn[1], cnz = cn[2];
    const float* gRow = guide + ((size_t)b * HH * WW + pix) * (K_SIZE * K_SIZE);

    float fw[SAMPLE_NUM];
    int   pls[SAMPLE_NUM];
#pragma unroll
    for (int s = 0; s < SAMPLE_NUM; ++s) {
        int pl = (lr + sDy[s]) * RCOLS + (lc + sDx[s]);
        pls[s] = pl;
        float vx = sV[pl];
        float dx0 = sN[pl * 4 + 0] - cnx;
        float dy0 = sN[pl * 4 + 1] - cny;
        float dz0 = sN[pl * 4 + 2] - cnz;
        float diff = sqrtf(dx0 * dx0 + dy0 * dy0 + dz0 * dz0);
        float nw = __expf(-0.5f * diff);
        fw[s] = vx * nw * gRow[sP[s]];
    }

    // softmax over samples (max-subtracted, matching jax.nn.softmax)
    float m = fw[0];
#pragma unroll
    for (int s = 1; s < SAMPLE_NUM; ++s) m = fmaxf(m, fw[s]);
    float sum = 0.0f;
#pragma unroll
    for (int s = 0; s < SAMPLE_NUM; ++s) { float e = __expf(fw[s] - m); fw[s] = e; sum += e; }
    const float inv = 1.0f / sum;

    vfloat4 acc[CC / 4];
#pragma unroll
    for (int j = 0; j < CC / 4; ++j) acc[j] = (vfloat4)(0.0f);

#pragma unroll
    for (int s = 0; s < SAMPLE_NUM; ++s) {
        float wgt = fw[s] * inv;
        const vfloat4* fp = (const vfloat4*)&sF[pls[s] * FSTRIDE];   // 16B aligned (144B stride)
#pragma unroll
        for (int j = 0; j < CC / 4; ++j) {
            vfloat4 f = fp[j];
            acc[j] += wgt * f;     // element-wise FMA on native vector
        }
    }

    const float* accS = (const float*)acc;
    float* outB = out + (size_t)b * CC * HH * WW;
#pragma unroll
    for (int c = 0; c < CC; ++c)                                   // coalesced across lanes
        __builtin_nontemporal_store(accS[c], &outB[(size_t)c * (HH * WW) + pix]);
}

__global__ __launch_bounds__(256) void
feat_copy_kernel(const vfloat4* __restrict__ src, vfloat4* __restrict__ dst, int n) {
    int i = blockIdx.x * blockDim.x + threadIdx.x;
    int stride = gridDim.x * blockDim.x;
    for (; i < n; i += stride) {
        vfloat4 v = __builtin_nontemporal_load(&src[i]);
        __builtin_nontemporal_store(v, &dst[i]);
    }
}

extern "C" void kernel_launch(void* const* d_in, const int* in_sizes, int n_in,
                              void* d_out, int out_size, void* d_ws, size_t ws_size,
                              hipStream_t stream) {
    (void)in_sizes; (void)n_in; (void)out_size; (void)d_ws; (void)ws_size;
    const float* depth = (const float*)d_in[0];
    const float* snorm = (const float*)d_in[1];
    const float* feat  = (const float*)d_in[2];
    const float* guide = (const float*)d_in[3];
    const int*   sidx  = (const int*)d_in[4];

    float* out  = (float*)d_out;
    float* out2 = out + (size_t)BB * CC * HH * WW;

    dim3 grid(WW / TW, HH / TH, BB);   // 10 x 30 x 2
    AdaptiveSample_main<<<grid, 256, SMEM_BYTES, stream>>>(depth, snorm, feat, guide, sidx, out);

    const int n4 = (BB * CC * HH * WW) / 4;   // 1,228,800 float4
    feat_copy_kernel<<<1200, 256, 0, stream>>>((const vfloat4*)feat, (vfloat4*)out2, n4);
}